// OTitansMemoryGate_60232621359108
// MI455X (gfx1250) — compile-verified
//
#include <hip/hip_runtime.h>
#include <hip/hip_bf16.h>

typedef float v2f __attribute__((ext_vector_type(2)));
typedef float v8f __attribute__((ext_vector_type(8)));
typedef unsigned int u32x4 __attribute__((ext_vector_type(4)));
typedef int i32x4 __attribute__((ext_vector_type(4)));
typedef int i32x8 __attribute__((ext_vector_type(8)));
typedef void __attribute__((address_space(3))) lds_void_t;

#define WMMA_F32(A_, B_, C_) \
  __builtin_amdgcn_wmma_f32_16x16x4_f32(false, (A_), false, (B_), (short)0, (C_), false, false)

constexpr int Bq = 4, Sq = 512, Hq = 512;
constexpr int T = 16, NCHUNK = Sq / T;

// powers of MOMENTUM=0.9, LPOW[i] = 0.9^i
__device__ const float LPOW[17] = {
    1.0f, 0.9f, 0.81f, 0.729f, 0.6561f, 0.59049f, 0.531441f, 0.4782969f,
    0.43046721f, 0.387420489f, 0.3486784401f, 0.31381059609f, 0.282429536481f,
    0.2541865828329f, 0.22876792454961f, 0.205891132094649f, 0.1853020188851841f};

// ---------------------------------------------------------------------------
// TDM: load a 2D f32 tile (rows x rowElems, row stride = rowElems) from global
// into LDS via TENSOR_LOAD_TO_LDS.  Descriptor per CDNA5 ISA ch.8:
//   group0: count=1 | lds_addr | global_addr[56:0] | type=2
//   group1: data_size=4B, tensor_dim0/1, tile_dim0/1, tensor_dim0_stride
//   group2/3: zero (2D tensor)
// 6-arg builtin form (clang-23 / therock-10.0 headers).
// Issue from ONE wave only (TDM ignores EXEC; suppression is by branch).
// ---------------------------------------------------------------------------
__device__ __forceinline__ void tdm_load_2d_f32(const float* gsrc, float* ldst,
                                                int rows, int rowElems) {
  const unsigned long long ga = (unsigned long long)(size_t)gsrc;
  const unsigned int lds =
      (unsigned int)(size_t)(lds_void_t*)ldst;  // LDS byte offset
  u32x4 g0 = {
      1u,                                            // count=1, user descriptor
      lds,                                           // lds_addr
      (unsigned int)(ga & 0xFFFFFFFFull),            // global_addr[31:0]
      (unsigned int)((ga >> 32) & 0x01FFFFFFull) |   // global_addr[56:32]
          (2u << 30)                                 // type = 2 ("image")
  };
  i32x8 g1 = {
      (int)(2u << 16),                               // data_size = 4 bytes
      (int)((rowElems & 0xFFFF) << 16),              // tensor_dim0[15:0]
      (int)(((rows & 0xFFFF) << 16) |                // tensor_dim1[15:0]
            ((rowElems >> 16) & 0xFFFF)),            // tensor_dim0[31:16]
      (int)(((rowElems & 0xFFFF) << 16) |            // tile_dim0
            ((rows >> 16) & 0xFFFF)),                // tensor_dim1[31:16]
      (int)(rows & 0xFFFF),                          // tile_dim1 (tile_dim2=0)
      (int)rowElems,                                 // tensor_dim0_stride[31:0]
      0,                                             // stride[47:32], dim1_stride lo
      0                                              // dim1_stride hi
  };
  i32x4 gz4 = {0, 0, 0, 0};
  i32x8 gz8 = {0, 0, 0, 0, 0, 0, 0, 0};
  __builtin_amdgcn_tensor_load_to_lds(g0, g1, gz4, gz4, gz8, 0);
}

// ---------------------------------------------------------------------------
// Kernel 1: q/k/v projections.  [2048,512] x [512,512], fp32 WMMA 16x16x4.
// grid = (128 row-tiles, 3 matrices), block = 256 (8 waves, 4 j-tiles each).
// ---------------------------------------------------------------------------
__global__ __launch_bounds__(256) void qkv_gemm(
    const float* __restrict__ X, const float* __restrict__ Wq,
    const float* __restrict__ Wk, const float* __restrict__ Wv,
    float* __restrict__ Q, float* __restrict__ K, float* __restrict__ V) {
  const int rowTile = blockIdx.x;
  const int which = blockIdx.y;
  const float* W = (which == 0) ? Wq : (which == 1) ? Wk : Wv;
  float* O = (which == 0) ? Q : (which == 1) ? K : V;

  const int lane = threadIdx.x & 31;
  const int wave = threadIdx.x >> 5;
  const int mrow = lane & 15;           // A row / B-C column within tile
  const int kh = (lane >> 4) << 1;      // K pair offset: 0 or 2

  const float* Xrow = X + (size_t)(rowTile * 16 + mrow) * Hq;

  v8f acc[4] = {};
  for (int k0 = 0; k0 < Hq; k0 += 4) {
    v2f a;
    a.x = Xrow[k0 + kh + 0];
    a.y = Xrow[k0 + kh + 1];
    if (k0 + 4 < Hq) __builtin_prefetch(Xrow + k0 + 4 + kh, 0, 3);  // near prefetch
#pragma unroll
    for (int t = 0; t < 4; ++t) {
      const int j0 = (wave * 4 + t) * 16;
      v2f b;
      b.x = W[(size_t)(k0 + kh + 0) * Hq + j0 + mrow];
      b.y = W[(size_t)(k0 + kh + 1) * Hq + j0 + mrow];
      acc[t] = WMMA_F32(a, b, acc[t]);
    }
  }
#pragma unroll
  for (int t = 0; t < 4; ++t) {
    const int j0 = (wave * 4 + t) * 16;
#pragma unroll
    for (int r = 0; r < 8; ++r) {
      const int mm = r + ((lane >> 4) << 3);
      const int nn = lane & 15;
      O[(size_t)(rowTile * 16 + mm) * Hq + j0 + nn] = acc[t][r];
    }
  }
}

// ---------------------------------------------------------------------------
// Kernel 2: chunked delta-rule scan.  One workgroup per batch (512 thr = 16
// waves), 32 chunks of T=16 steps.  Per chunk:
//   P0: TDM tensor_load_to_lds stages K_c and V_c tiles (32 KB each)
//   P1: KM = K_c@M, QM = Q_c@M   (WMMA, M read once, shared B-frags)
//   P2: forward substitution for surprise rows S + 16x16 dot matrices
//   P3: retrieval = lam^t*QM + corrections  -> mem_out
//   P4: M <- lam^16*M + S'^T K_c  (WMMA, K-dim 16)
// Dynamic LDS: KM,QM,Kc,S (4 x 32 KB) + two 16x16 mats  = 130 KB (< 320 KB).
// ---------------------------------------------------------------------------
__global__ __launch_bounds__(512) void titans_scan(
    const float* __restrict__ Q, const float* __restrict__ K,
    const float* __restrict__ V, float* __restrict__ M,
    float* __restrict__ MEM) {
  extern __shared__ float smem[];
  float* sKM = smem;                 // [T][Hq]
  float* sQM = sKM + T * Hq;         // [T][Hq]
  float* sKc = sQM + T * Hq;         // [T][Hq]
  float* sS  = sKc + T * Hq;         // [T][Hq]  (init = V chunk, becomes surprise)
  float* sAK = sS + T * Hq;          // [T][T]  k_t . s_u
  float* sAQ = sAK + T * T;          // [T][T]  q_t . s_u

  const int b = blockIdx.x;
  const int tid = threadIdx.x;
  const int lane = tid & 31;
  const int wave = tid >> 5;         // 0..15
  const int mrow = lane & 15;
  const int kh = (lane >> 4) << 1;

  float* Mb = M + (size_t)b * Hq * Hq;
  const float* Qb = Q + (size_t)b * Sq * Hq;
  const float* Kb = K + (size_t)b * Sq * Hq;
  const float* Vb = V + (size_t)b * Sq * Hq;
  float* MEMb = MEM + (size_t)b * Sq * Hq;

  // M starts at zero
  for (int i = tid; i < Hq * Hq; i += 512) Mb[i] = 0.f;
  __threadfence();
  __syncthreads();

  for (int c = 0; c < NCHUNK; ++c) {
    const int s0 = c * T;

    // ---- Phase 0: stage K chunk and V chunk (S init) via the Tensor Data
    // Mover.  Wave 0 issues both DMAs and waits on TENSORcnt; the workgroup
    // barrier releases the other 15 waves once LDS holds the tiles.
    if (wave == 0) {
      tdm_load_2d_f32(Kb + (size_t)s0 * Hq, sKc, T, Hq);
      tdm_load_2d_f32(Vb + (size_t)s0 * Hq, sS, T, Hq);
      __builtin_amdgcn_s_wait_tensorcnt(0);
    }
    __syncthreads();

    // ---- Phase 1: KM / QM via WMMA; each wave owns 2 j-tiles; M B-frags
    // are loaded once and feed both the K and Q accumulators.
    {
      const int j0 = (wave * 2 + 0) * 16, j1 = (wave * 2 + 1) * 16;
      v8f km0 = {}, km1 = {}, qm0 = {}, qm1 = {};
      const float* Krow = Kb + (size_t)(s0 + mrow) * Hq;
      const float* Qrow = Qb + (size_t)(s0 + mrow) * Hq;
      for (int k0 = 0; k0 < Hq; k0 += 4) {
        v2f ak, aq, b0, b1;
        ak.x = Krow[k0 + kh]; ak.y = Krow[k0 + kh + 1];
        aq.x = Qrow[k0 + kh]; aq.y = Qrow[k0 + kh + 1];
        b0.x = Mb[(size_t)(k0 + kh) * Hq + j0 + mrow];
        b0.y = Mb[(size_t)(k0 + kh + 1) * Hq + j0 + mrow];
        b1.x = Mb[(size_t)(k0 + kh) * Hq + j1 + mrow];
        b1.y = Mb[(size_t)(k0 + kh + 1) * Hq + j1 + mrow];
        km0 = WMMA_F32(ak, b0, km0);
        km1 = WMMA_F32(ak, b1, km1);
        qm0 = WMMA_F32(aq, b0, qm0);
        qm1 = WMMA_F32(aq, b1, qm1);
      }
#pragma unroll
      for (int r = 0; r < 8; ++r) {
        const int mm = r + ((lane >> 4) << 3);
        const int nn = lane & 15;
        sKM[mm * Hq + j0 + nn] = km0[r];
        sKM[mm * Hq + j1 + nn] = km1[r];
        sQM[mm * Hq + j0 + nn] = qm0[r];
        sQM[mm * Hq + j1 + nn] = qm1[r];
      }
    }
    __syncthreads();

    // ---- Phase 2: forward substitution (sequential in t; parallel in j)
    for (int t = 0; t < T; ++t) {
      float s = sS[t * Hq + tid] - LPOW[t] * sKM[t * Hq + tid];
      for (int u = 0; u < t; ++u)
        s -= LPOW[t - 1 - u] * sAK[t * T + u] * sKc[u * Hq + tid];
      sS[t * Hq + tid] = s;
      __syncthreads();
      // wave w computes k_w . s_t and q_w . s_t (wave32 shuffle reduce)
      {
        float pk = 0.f, pq = 0.f;
        const float* Qrow = Qb + (size_t)(s0 + wave) * Hq;
        for (int j = lane; j < Hq; j += 32) {
          const float sv = sS[t * Hq + j];
          pk += sKc[wave * Hq + j] * sv;
          pq += Qrow[j] * sv;
        }
#pragma unroll
        for (int off = 16; off > 0; off >>= 1) {
          pk += __shfl_xor(pk, off, 32);
          pq += __shfl_xor(pq, off, 32);
        }
        if (lane == 0) { sAK[wave * T + t] = pk; sAQ[wave * T + t] = pq; }
      }
      __syncthreads();
    }

    // ---- Phase 3: retrieval rows -> mem_out
    for (int t = 0; t < T; ++t) {
      float r = LPOW[t] * sQM[t * Hq + tid];
      for (int u = 0; u < t; ++u)
        r += LPOW[t - 1 - u] * sAQ[t * T + u] * sKc[u * Hq + tid];
      MEMb[(size_t)(s0 + t) * Hq + tid] = r;
    }
    __syncthreads();

    // ---- Phase 4: M <- lam^T * M + sum_u lam^{T-1-u} s_u k_u^T  (WMMA, K=16)
    for (int tile = wave; tile < 32 * 32; tile += 16) {
      const int it = (tile >> 5) * 16;
      const int jt = (tile & 31) * 16;
      v8f acc;
#pragma unroll
      for (int r = 0; r < 8; ++r) {
        const int mm = r + ((lane >> 4) << 3);
        const int nn = lane & 15;
        acc[r] = LPOW[T] * Mb[(size_t)(it + mm) * Hq + jt + nn];
      }
#pragma unroll
      for (int u0 = 0; u0 < T; u0 += 4) {
        v2f a, bb;
        a.x = LPOW[T - 1 - (u0 + kh)]     * sS[(u0 + kh) * Hq + it + mrow];
        a.y = LPOW[T - 1 - (u0 + kh + 1)] * sS[(u0 + kh + 1) * Hq + it + mrow];
        bb.x = sKc[(u0 + kh) * Hq + jt + mrow];
        bb.y = sKc[(u0 + kh + 1) * Hq + jt + mrow];
        acc = WMMA_F32(a, bb, acc);
      }
#pragma unroll
      for (int r = 0; r < 8; ++r) {
        const int mm = r + ((lane >> 4) << 3);
        const int nn = lane & 15;
        Mb[(size_t)(it + mm) * Hq + jt + nn] = acc[r];
      }
    }
    __threadfence();
    __syncthreads();
  }
}

// ---------------------------------------------------------------------------
// Kernel 3: gating MLP + residual.  grid = 128 row-tiles, block = 256.
// GEMM1 (K=1024, A = [hs|mem]) -> silu -> LDS, GEMM2 (K=128) -> sigmoid.
// ---------------------------------------------------------------------------
__global__ __launch_bounds__(256) void gate_mlp(
    const float* __restrict__ HS, const float* __restrict__ MEM,
    const float* __restrict__ W1, const float* __restrict__ b1,
    const float* __restrict__ W2, const float* __restrict__ b2,
    float* __restrict__ OUT) {
  __shared__ float sH[16 * 128];   // silu(h) tile, 8 KB
  const int rowTile = blockIdx.x;
  const int lane = threadIdx.x & 31;
  const int wave = threadIdx.x >> 5;   // 0..7
  const int mrow = lane & 15;
  const int kh = (lane >> 4) << 1;
  const int row0 = rowTile * 16;

  const float* HSrow = HS + (size_t)(row0 + mrow) * Hq;
  const float* MEMrow = MEM + (size_t)(row0 + mrow) * Hq;

  // GEMM1: h[16][128]; wave w owns columns w*16..w*16+15
  {
    const int j0 = wave * 16;
    v8f acc = {};
    for (int k0 = 0; k0 < 2 * Hq; k0 += 4) {
      const int kk = k0 + kh;
      v2f a, bb;
      a.x = (kk < Hq) ? HSrow[kk] : MEMrow[kk - Hq];
      a.y = (kk + 1 < Hq) ? HSrow[kk + 1] : MEMrow[kk + 1 - Hq];
      bb.x = W1[(size_t)kk * 128 + j0 + mrow];
      bb.y = W1[(size_t)(kk + 1) * 128 + j0 + mrow];
      acc = WMMA_F32(a, bb, acc);
    }
#pragma unroll
    for (int r = 0; r < 8; ++r) {
      const int mm = r + ((lane >> 4) << 3);
      const int nn = lane & 15;
      float h = acc[r] + b1[j0 + nn];
      sH[mm * 128 + j0 + nn] = h * (1.f / (1.f + __expf(-h)));  // silu
    }
  }
  __syncthreads();

  // GEMM2: gate[16][512]; wave owns 4 j-tiles; then residual mix.
  const float* sHrow = sH + mrow * 128;
#pragma unroll
  for (int tt = 0; tt < 4; ++tt) {
    const int j0 = (wave * 4 + tt) * 16;
    v8f acc = {};
    for (int k0 = 0; k0 < 128; k0 += 4) {
      v2f a, bb;
      a.x = sHrow[k0 + kh];
      a.y = sHrow[k0 + kh + 1];
      bb.x = W2[(size_t)(k0 + kh) * Hq + j0 + mrow];
      bb.y = W2[(size_t)(k0 + kh + 1) * Hq + j0 + mrow];
      acc = WMMA_F32(a, bb, acc);
    }
#pragma unroll
    for (int r = 0; r < 8; ++r) {
      const int mm = r + ((lane >> 4) << 3);
      const int nn = lane & 15;
      const size_t idx = (size_t)(row0 + mm) * Hq + j0 + nn;
      float g = 1.f / (1.f + __expf(-(acc[r] + b2[j0 + nn])));
      OUT[idx] = HS[idx] + g * MEM[idx];
    }
  }
}

// ---------------------------------------------------------------------------
extern "C" void kernel_launch(void* const* d_in, const int* in_sizes, int n_in,
                              void* d_out, int out_size, void* d_ws,
                              size_t ws_size, hipStream_t stream) {
  const float* hs = (const float*)d_in[0];
  const float* Wq = (const float*)d_in[1];
  const float* Wk = (const float*)d_in[2];
  const float* Wv = (const float*)d_in[3];
  const float* W1 = (const float*)d_in[4];
  const float* b1 = (const float*)d_in[5];
  const float* W2 = (const float*)d_in[6];
  const float* b2 = (const float*)d_in[7];
  float* out = (float*)d_out;

  // workspace layout (floats): q | k | v | M | mem  = 5 * 4 MB = 20 MB
  float* ws = (float*)d_ws;
  const size_t NBSH = (size_t)Bq * Sq * Hq;
  float* q = ws;
  float* k = q + NBSH;
  float* v = k + NBSH;
  float* M = v + NBSH;                  // [B][H][H]
  float* mem = M + (size_t)Bq * Hq * Hq;

  qkv_gemm<<<dim3(Bq * Sq / 16, 3), 256, 0, stream>>>(hs, Wq, Wk, Wv, q, k, v);

  const size_t shBytes = (size_t)(4 * T * Hq + 2 * T * T) * sizeof(float); // 130 KB
  titans_scan<<<dim3(Bq), 512, shBytes, stream>>>(q, k, v, M, mem);

  gate_mlp<<<dim3(Bq * Sq / 16), 256, 0, stream>>>(hs, mem, W1, b1, W2, b2, out);
}